// SplitRoundGIN_noparam_29257317220550
// MI455X (gfx1250) — compile-verified
//
#include <hip/hip_runtime.h>

// ---------------------------------------------------------------------------
// SplitRoundGIN on MI455X (gfx1250, wave32).
//   - GIN with eps=-1  ==> pure neighbor scatter-sum (no self term).
//   - LayerNorm reductions over D=64 done with V_WMMA_F32_16X16X4_F32:
//     16 chained f32 WMMAs vs an all-ones B give 16 nodes' row-sums
//     (ones-B is layout agnostic); a second accumulator on A*A gives sum(x^2).
//   - Edge scatter: one wave per edge, 32 lanes x float2 = D=64,
//     hardware global_atomic_add_f32 (L2-resident: working set << 192MB L2).
// ---------------------------------------------------------------------------

typedef float v2f __attribute__((ext_vector_type(2)));
typedef float v8f __attribute__((ext_vector_type(8)));

#define DF 64  // feature dim (reference D)

__device__ __forceinline__ float2 ld2(const float* p) { return *(const float2*)p; }
__device__ __forceinline__ void st2(float* p, float2 v) { *(float2*)p = v; }

// ---------------------------------------------------------------------------
// LayerNorm (elementwise_affine=False) with optional ReLU, 16 nodes per wave.
// A-matrix layout for V_WMMA_F32_16X16X4_F32 (ISA 7.12.2):
//   lane L: M = L%16;  VGPR0/1 hold K = 2*(L/16) + {0,1}
// C/D layout: VGPR r holds M = r + 8*(L/16), replicated across columns.
// ---------------------------------------------------------------------------
template <bool RELU>
__global__ void ln_wmma_kernel(const float* __restrict__ in,
                               float* __restrict__ out, int nWaves) {
  const int lane  = threadIdx.x & 31;
  const int gwave = blockIdx.x * (blockDim.x >> 5) + (threadIdx.x >> 5);
  if (gwave >= nWaves) return;  // wave-uniform: EXEC stays all-1s for WMMA

  const int m    = lane & 15;
  const int half = lane >> 4;
  const size_t base = ((size_t)gwave * 16 + m) * DF + (size_t)half * 2;
  const float* row = in + base;

  v2f vals[16];
  v8f s = {0.f, 0.f, 0.f, 0.f, 0.f, 0.f, 0.f, 0.f};
  v8f q = {0.f, 0.f, 0.f, 0.f, 0.f, 0.f, 0.f, 0.f};
  const v2f ones = {1.0f, 1.0f};

#pragma unroll
  for (int j = 0; j < 16; ++j) {
    v2f a = *(const v2f*)(row + j * 4);  // feats 4j + 2*half + {0,1}
    vals[j] = a;
    // D = A x Ones + C : accumulates per-node feature sums (f32, exact layout
    // of B irrelevant since every element is 1.0).
    s = __builtin_amdgcn_wmma_f32_16x16x4_f32(false, a, false, ones,
                                              (short)0, s, false, false);
    v2f a2 = a * a;
    q = __builtin_amdgcn_wmma_f32_16x16x4_f32(false, a2, false, ones,
                                              (short)0, q, false, false);
  }

  // Node m's sums live in VGPR (m&7) of the lane-half (m>>3).
  float sum_a = s[m & 7];
  float ssq_a = q[m & 7];
  float sum_b = __shfl_xor(sum_a, 16, 32);
  float ssq_b = __shfl_xor(ssq_a, 16, 32);
  const bool mine = ((m >> 3) == half);
  float sum = mine ? sum_a : sum_b;
  float ssq = mine ? ssq_a : ssq_b;

  const float mean = sum * (1.0f / 64.0f);
  const float var  = fmaxf(ssq * (1.0f / 64.0f) - mean * mean, 0.0f);
  const float inv  = rsqrtf(var + 1e-5f);

  float* orow = out + base;
#pragma unroll
  for (int j = 0; j < 16; ++j) {
    v2f v = (vals[j] - mean) * inv;
    if (RELU) {
      v.x = fmaxf(v.x, 0.0f);
      v.y = fmaxf(v.y, 0.0f);
    }
    *(v2f*)(orow + j * 4) = v;
  }
}

// ---------------------------------------------------------------------------
// Zero a node buffer (float4 stores).
// ---------------------------------------------------------------------------
__global__ void zero_kernel(float4* __restrict__ p, long total4) {
  long i = (long)blockIdx.x * blockDim.x + threadIdx.x;
  long stride = (long)gridDim.x * blockDim.x;
  const float4 z = make_float4(0.f, 0.f, 0.f, 0.f);
  for (; i < total4; i += stride) p[i] = z;
}

// ---------------------------------------------------------------------------
// Edge scatter-sum: agg[dst] += h[src] (optionally masked by age[src] >= w).
// One wave per edge: lane holds float2 -> 32 lanes cover D=64 exactly.
// Mask is wave-uniform, so masked edges skip gather + atomics entirely.
// ---------------------------------------------------------------------------
template <bool MASKED>
__global__ void scatter_kernel(const float* __restrict__ h,
                               float* __restrict__ agg,
                               const int* __restrict__ src,
                               const int* __restrict__ dst,
                               const int* __restrict__ age, int w, int E) {
  const int lane   = threadIdx.x & 31;
  const int wave   = blockIdx.x * (blockDim.x >> 5) + (threadIdx.x >> 5);
  const int stride = gridDim.x * (blockDim.x >> 5);
  const int c = lane * 2;
  for (int e = wave; e < E; e += stride) {
    // Stream the edge lists ahead (global_prefetch_b8; speculative, safe OOB).
    __builtin_prefetch(src + e + stride, 0, 0);
    __builtin_prefetch(dst + e + stride, 0, 0);
    const int sN = src[e];
    if (MASKED) {
      if (age[sN] < w) continue;  // uniform across wave
    }
    const int dN = dst[e];
    float2 v = ld2(h + (size_t)sN * DF + c);
    float* p = agg + (size_t)dN * DF + c;
    unsafeAtomicAdd(p, v.x);       // global_atomic_add_f32 (L2-resident)
    unsafeAtomicAdd(p + 1, v.y);
  }
}

// ---------------------------------------------------------------------------
// Window 0 writeout: out[:,0:192] = [hn|h1|h2]; out[:,192:384] = same
// (h_temp starts as windows[0]; windows 1,2 subtract 0.5x later).
// Covers every output element -> no separate init needed.
// ---------------------------------------------------------------------------
__global__ void writeout0_kernel(const float* __restrict__ hn,
                                 const float* __restrict__ h1,
                                 const float* __restrict__ h2,
                                 float* __restrict__ out, int N) {
  int tid = blockIdx.x * blockDim.x + threadIdx.x;
  if (tid >= N * 32) return;
  int n = tid >> 5;
  int c = (tid & 31) * 2;
  size_t fb = (size_t)n * DF + c;
  float2 a = ld2(hn + fb), b = ld2(h1 + fb), g = ld2(h2 + fb);
  float* o = out + (size_t)n * 384;
  st2(o + c, a);        st2(o + 64 + c, b);   st2(o + 128 + c, g);
  st2(o + 192 + c, a);  st2(o + 256 + c, b);  st2(o + 320 + c, g);
}

// ---------------------------------------------------------------------------
// Windows 1,2: out[:,192:384] -= 0.5 * [hn*mask | h1 | h2]
// (feats[0] of window w is the age-masked hn; h1/h2 already embed the mask
//  through the masked layer-1 aggregation).
// ---------------------------------------------------------------------------
__global__ void writeoutw_kernel(const float* __restrict__ hn,
                                 const float* __restrict__ h1,
                                 const float* __restrict__ h2,
                                 const int* __restrict__ age,
                                 float* __restrict__ out, int w, int N) {
  int tid = blockIdx.x * blockDim.x + threadIdx.x;
  if (tid >= N * 32) return;
  int n = tid >> 5;
  int c = (tid & 31) * 2;
  size_t fb = (size_t)n * DF + c;
  const float coefA = (age[n] >= w) ? 0.5f : 0.0f;
  float2 a = ld2(hn + fb), b = ld2(h1 + fb), g = ld2(h2 + fb);
  float* o = out + (size_t)n * 384;
  float2 t;
  t = ld2(o + 192 + c); t.x -= coefA * a.x; t.y -= coefA * a.y; st2(o + 192 + c, t);
  t = ld2(o + 256 + c); t.x -= 0.5f * b.x;  t.y -= 0.5f * b.y;  st2(o + 256 + c, t);
  t = ld2(o + 320 + c); t.x -= 0.5f * g.x;  t.y -= 0.5f * g.y;  st2(o + 320 + c, t);
}

// ---------------------------------------------------------------------------
extern "C" void kernel_launch(void* const* d_in, const int* in_sizes, int n_in,
                              void* d_out, int out_size, void* d_ws,
                              size_t ws_size, hipStream_t stream) {
  const float* feature = (const float*)d_in[0];
  const int* age = (const int*)d_in[1];
  const int* src = (const int*)d_in[2];
  const int* dst = (const int*)d_in[3];
  float* out = (float*)d_out;

  const int N = in_sizes[1];  // 100000
  const int E = in_sizes[2];  // 1200000

  // Scratch: hn | bufA | bufB  (3 * N * 64 floats = 76.8 MB)
  float* hn   = (float*)d_ws;
  float* bufA = hn + (size_t)N * DF;
  float* bufB = bufA + (size_t)N * DF;

  const int nWaves   = N / 16;                 // 6250 LN waves (N%16==0)
  const int lnBlocks = (nWaves + 1) / 2;       // 64-thread (2-wave) blocks
  const long total4  = (long)N * (DF / 4);
  const int woBlocks = (N * 32 + 255) / 256;

  // Input LayerNorm.
  ln_wmma_kernel<false><<<lnBlocks, 64, 0, stream>>>(feature, hn, nWaves);

  for (int w = 0; w < 3; ++w) {
    // Layer 1: agg = scatter(hn * (age>=w)); h1 = relu(LN(agg)) in place.
    zero_kernel<<<2048, 256, 0, stream>>>((float4*)bufA, total4);
    if (w == 0)
      scatter_kernel<false><<<1024, 256, 0, stream>>>(hn, bufA, src, dst, age, w, E);
    else
      scatter_kernel<true><<<1024, 256, 0, stream>>>(hn, bufA, src, dst, age, w, E);
    ln_wmma_kernel<true><<<lnBlocks, 64, 0, stream>>>(bufA, bufA, nWaves);

    // Layer 2: agg = scatter(h1); h2 = relu(LN(agg)) in place.
    zero_kernel<<<2048, 256, 0, stream>>>((float4*)bufB, total4);
    scatter_kernel<false><<<1024, 256, 0, stream>>>(bufA, bufB, src, dst, age, 0, E);
    ln_wmma_kernel<true><<<lnBlocks, 64, 0, stream>>>(bufB, bufB, nWaves);

    // Accumulate into output.
    if (w == 0)
      writeout0_kernel<<<woBlocks, 256, 0, stream>>>(hn, bufA, bufB, out, N);
    else
      writeoutw_kernel<<<woBlocks, 256, 0, stream>>>(hn, bufA, bufB, age, out, w, N);
  }
}